// SiluAndMul_5334349382092
// MI455X (gfx1250) — compile-verified
//
#include <hip/hip_runtime.h>
#include <hip/hip_bf16.h>
#include <cstdint>

// SiLU-and-mul: out[r, c] = silu(x[r, c]) * x[r, d + c],  d = 16384
// x: f32[8192 rows, 32768], out: f32[8192 rows, 16384]
//
// Pure streaming kernel: 1.5 GiB of traffic, ~69us floor at 23.3 TB/s.
// Strategy: b128 accesses + non-temporal cache hints (no reuse; stream is
// 8x the 192MB L2), wave32-friendly 256-thread blocks, and hardware
// v_rcp_f32 / v_exp_f32 to keep VALU issue well under the HBM service rate.

typedef __attribute__((ext_vector_type(4))) float v4f;

// Shape constants for this problem (d = 16384 floats).
#define DV4_SHIFT   12      // 4096 float4 per output row  (= d/4)
#define DV4_MASK    4095
#define INROW_V4    8192    // 2*d/4 float4 per input row
#define OUTROW_V4   4096

__global__ __launch_bounds__(256) void silu_mul_f32_kernel(
    const float* __restrict__ x, float* __restrict__ out, int nvec) {
  int idx = blockIdx.x * blockDim.x + threadIdx.x;
  if (idx >= nvec) return;

  int r = idx >> DV4_SHIFT;      // output row
  int c = idx & DV4_MASK;        // float4 column within row

  const v4f* __restrict__ xv = (const v4f*)x;
  size_t base = (size_t)r * INROW_V4 + (size_t)c;

  // Non-temporal b128 loads: gate (first half of row), up (second half).
  v4f g = __builtin_nontemporal_load(xv + base);
  v4f u = __builtin_nontemporal_load(xv + base + OUTROW_V4);

  v4f o;
#pragma unroll
  for (int k = 0; k < 4; ++k) {
    float gk = g[k];
    // silu(g) = g * sigmoid(g) = g * rcp(1 + exp(-g))
    // v_exp_f32 + v_rcp_f32: both single-instruction trans-pipe ops (~1 ulp).
    float s = __builtin_amdgcn_rcpf(1.0f + __expf(-gk));
    o[k] = gk * s * u[k];
  }

  // Non-temporal b128 store: output is written once, never re-read.
  __builtin_nontemporal_store(o, (v4f*)out + idx);
}

extern "C" void kernel_launch(void* const* d_in, const int* in_sizes, int n_in,
                              void* d_out, int out_size, void* d_ws, size_t ws_size,
                              hipStream_t stream) {
  (void)in_sizes; (void)n_in; (void)d_ws; (void)ws_size;

  const float* x = (const float*)d_in[0];
  float* out = (float*)d_out;

  // out_size = 4*2048*16384 = 134,217,728 floats -> 33,554,432 float4s.
  int nvec = out_size / 4;

  const int block = 256;                    // 8 wave32 per block
  int grid = (nvec + block - 1) / block;    // 131,072 blocks

  silu_mul_f32_kernel<<<grid, block, 0, stream>>>(x, out, nvec);
}